// ConvNeXtBlock_Sparse_70944269795965
// MI455X (gfx1250) — compile-verified
//
#include <hip/hip_runtime.h>
#include <math.h>

// Problem constants (from reference)
#define NPTS 200000
#define DIM  128
#define KOFF 49
#define HID  512

// Tiling
#define MT   64                    // rows per workgroup (200000 = 3125 * 64)
#define HSTR (DIM + 4)             // padded LDS stride for LN tile (132 floats)
#define TSTR (HID + 4)             // padded LDS stride for GELU tile (516 floats)
#define LDS_BYTES ((MT * HSTR + MT * TSTR) * 4)   // 165,888 B < 320 KB WGP LDS

typedef float v2f __attribute__((ext_vector_type(2)));
typedef float v8f __attribute__((ext_vector_type(8)));

// ---------------------------------------------------------------------------
// Kernel 1: acc[n][c] = b_dw[c]   (fold depthwise bias into the accumulator)
// ---------------------------------------------------------------------------
__global__ void init_acc_kernel(const float* __restrict__ b_dw,
                                float* __restrict__ acc) {
  int i = blockIdx.x * blockDim.x + threadIdx.x;       // float4 index
  if (i >= NPTS * DIM / 4) return;
  ((float4*)acc)[i] = ((const float4*)b_dw)[i & (DIM / 4 - 1)];
}

// ---------------------------------------------------------------------------
// Kernel 2: sparse depthwise conv scatter-add.
// One wave32 per (k, n) pair; each lane handles 4 channels (float4 gather,
// 4x global_atomic_add_f32 scatter). x is L2-resident (102MB < 192MB L2).
// ---------------------------------------------------------------------------
__global__ void dwconv_scatter_kernel(const float* __restrict__ x,
                                      const int* __restrict__ in_maps,
                                      const int* __restrict__ out_maps,
                                      const float* __restrict__ w_dw,
                                      float* __restrict__ acc) {
  unsigned tid  = blockIdx.x * blockDim.x + threadIdx.x;
  unsigned wave = tid >> 5;                       // flat (k*N + n)
  int lane = threadIdx.x & 31;
  if (wave >= (unsigned)(KOFF * NPTS)) return;
  int k = (int)(wave / NPTS);
  int in_idx  = in_maps[wave];
  int out_idx = out_maps[wave];
  float4 xv = ((const float4*)(x    + (size_t)in_idx * DIM))[lane];
  float4 wv = ((const float4*)(w_dw + (size_t)k      * DIM))[lane];
  float* dst = acc + (size_t)out_idx * DIM + lane * 4;
  unsafeAtomicAdd(dst + 0, xv.x * wv.x);
  unsafeAtomicAdd(dst + 1, xv.y * wv.y);
  unsafeAtomicAdd(dst + 2, xv.z * wv.z);
  unsafeAtomicAdd(dst + 3, xv.w * wv.w);
}

// ---------------------------------------------------------------------------
// Kernel 3: fused LayerNorm -> GEMM1(+GELU) -> GEMM2 -> layerscale+residual.
// 256 threads = 8 waves per 64-row tile. fp32 WMMA 16x16x4 for both GEMMs.
// ---------------------------------------------------------------------------
__device__ __forceinline__ float gelu_exact(float v) {
  return 0.5f * v * (1.0f + erff(v * 0.70710678118654752440f));
}

__global__ __launch_bounds__(256)
void ln_mlp_fused_kernel(const float* __restrict__ x,
                         const float* __restrict__ acc,
                         const float* __restrict__ ln_g,
                         const float* __restrict__ ln_b,
                         const float* __restrict__ w1,
                         const float* __restrict__ b1,
                         const float* __restrict__ w2,
                         const float* __restrict__ b2,
                         const float* __restrict__ gamma,
                         float* __restrict__ out) {
  extern __shared__ float smem[];
  float* sH = smem;                 // [MT][HSTR] : LayerNorm output
  float* sT = smem + MT * HSTR;     // [MT][TSTR] : GELU(h@w1+b1)

  const int wid  = threadIdx.x >> 5;   // 0..7
  const int lane = threadIdx.x & 31;
  const int rowBase = blockIdx.x * MT;

  // ---- Phase 1: LayerNorm over channels, write normalized tile to LDS ----
  for (int r = wid; r < MT; r += 8) {
    float4 v = ((const float4*)(acc + (size_t)(rowBase + r) * DIM))[lane];
    float s  = v.x + v.y + v.z + v.w;
    float s2 = v.x * v.x + v.y * v.y + v.z * v.z + v.w * v.w;
    #pragma unroll
    for (int off = 16; off > 0; off >>= 1) {
      s  += __shfl_xor(s,  off, 32);
      s2 += __shfl_xor(s2, off, 32);
    }
    float mean = s * (1.0f / DIM);
    float var  = s2 * (1.0f / DIM) - mean * mean;
    float rstd = rsqrtf(var + 1e-6f);
    float4 g = ((const float4*)ln_g)[lane];
    float4 b = ((const float4*)ln_b)[lane];
    float* drow = sH + r * HSTR + lane * 4;
    drow[0] = (v.x - mean) * rstd * g.x + b.x;
    drow[1] = (v.y - mean) * rstd * g.y + b.y;
    drow[2] = (v.z - mean) * rstd * g.z + b.z;
    drow[3] = (v.w - mean) * rstd * g.w + b.w;
  }
  __syncthreads();

  // Common WMMA lane decomposition (ISA fp32 16x16x4 layouts):
  //  A 16x4 : lanes 0-15 rows M, VGPR0/1 = K0/K1 ; lanes 16-31 = K2/K3
  //  B 4x16 : lanes 0-15 cols N, VGPR0/1 = K0/K1 ; lanes 16-31 = K2/K3
  //  C 16x16: VGPR j = row (j | half*8), col = lane&15
  const int mrow  = lane & 15;
  const int khalf = (lane >> 4) << 1;   // 0 or 2
  const int chalf = (lane >> 4) << 3;   // 0 or 8

  // ---- Phase 2: GEMM1  sT = GELU(sH(64x128) @ w1(128x512) + b1) ----
  {
    const int n0 = wid * 64;            // this wave's 64-column slice of HID
    v8f zero = {};
    v8f C[4][4];
    #pragma unroll
    for (int mt = 0; mt < 4; ++mt)
      #pragma unroll
      for (int nt = 0; nt < 4; ++nt) C[mt][nt] = zero;

    for (int kk = 0; kk < DIM; kk += 4) {
      const int kA = kk + khalf;
      v2f A[4], B[4];
      #pragma unroll
      for (int mt = 0; mt < 4; ++mt) {
        const float* p = sH + (mt * 16 + mrow) * HSTR + kA;
        A[mt].x = p[0]; A[mt].y = p[1];
      }
      #pragma unroll
      for (int nt = 0; nt < 4; ++nt) {
        const float* p = w1 + (size_t)kA * HID + (n0 + nt * 16 + mrow);
        B[nt].x = p[0]; B[nt].y = p[HID];
      }
      #pragma unroll
      for (int mt = 0; mt < 4; ++mt)
        #pragma unroll
        for (int nt = 0; nt < 4; ++nt)
          C[mt][nt] = __builtin_amdgcn_wmma_f32_16x16x4_f32(
              false, A[mt], false, B[nt], (short)0, C[mt][nt], false, false);
    }
    #pragma unroll
    for (int mt = 0; mt < 4; ++mt)
      #pragma unroll
      for (int nt = 0; nt < 4; ++nt) {
        const int col  = n0 + nt * 16 + mrow;
        const float bb = b1[col];
        #pragma unroll
        for (int j = 0; j < 8; ++j) {
          const int row = mt * 16 + chalf + j;
          sT[row * TSTR + col] = gelu_exact(C[mt][nt][j] + bb);
        }
      }
  }
  __syncthreads();

  // ---- Phase 3: GEMM2  o = sT(64x512) @ w2(512x128), fuse b2/gamma/residual ----
  {
    const int c0 = wid * 16;            // this wave's 16-column slice of DIM
    v8f zero = {};
    v8f D[4];
    #pragma unroll
    for (int mt = 0; mt < 4; ++mt) D[mt] = zero;

    for (int kk = 0; kk < HID; kk += 4) {
      const int kA = kk + khalf;
      v2f A[4], B;
      #pragma unroll
      for (int mt = 0; mt < 4; ++mt) {
        const float* p = sT + (mt * 16 + mrow) * TSTR + kA;
        A[mt].x = p[0]; A[mt].y = p[1];
      }
      const float* q = w2 + (size_t)kA * DIM + (c0 + mrow);
      B.x = q[0]; B.y = q[DIM];
      #pragma unroll
      for (int mt = 0; mt < 4; ++mt)
        D[mt] = __builtin_amdgcn_wmma_f32_16x16x4_f32(
            false, A[mt], false, B, (short)0, D[mt], false, false);
    }

    const int col  = c0 + mrow;
    const float bb = b2[col];
    const float gm = gamma[col];
    #pragma unroll
    for (int mt = 0; mt < 4; ++mt)
      #pragma unroll
      for (int j = 0; j < 8; ++j) {
        const int row = rowBase + mt * 16 + chalf + j;
        const size_t idx = (size_t)row * DIM + col;
        out[idx] = x[idx] + gm * (D[mt][j] + bb);
      }
  }
}

// ---------------------------------------------------------------------------
extern "C" void kernel_launch(void* const* d_in, const int* in_sizes, int n_in,
                              void* d_out, int out_size, void* d_ws, size_t ws_size,
                              hipStream_t stream) {
  const float* x       = (const float*)d_in[0];
  const int*   in_maps = (const int*)  d_in[1];
  const int*   out_maps= (const int*)  d_in[2];
  const float* w_dw    = (const float*)d_in[3];
  const float* b_dw    = (const float*)d_in[4];
  const float* ln_g    = (const float*)d_in[5];
  const float* ln_b    = (const float*)d_in[6];
  const float* w1      = (const float*)d_in[7];
  const float* b1      = (const float*)d_in[8];
  const float* w2      = (const float*)d_in[9];
  const float* b2      = (const float*)d_in[10];
  const float* gamma   = (const float*)d_in[11];
  float* out = (float*)d_out;
  float* acc = (float*)d_ws;   // needs N*DIM*4 = 102.4 MB scratch

  (void)in_sizes; (void)n_in; (void)out_size; (void)ws_size;

  // 1) acc = broadcast(b_dw)
  init_acc_kernel<<<(NPTS * DIM / 4 + 255) / 256, 256, 0, stream>>>(b_dw, acc);

  // 2) scatter-add depthwise conv (one wave per (k,n))
  const unsigned nwaves  = (unsigned)KOFF * (unsigned)NPTS;       // 9.8M waves
  const unsigned nthread = nwaves * 32u;                          // 313.6M threads
  dwconv_scatter_kernel<<<(nthread + 255u) / 256u, 256, 0, stream>>>(
      x, in_maps, out_maps, w_dw, acc);

  // 3) fused LN + MLP + layerscale + residual (fp32 WMMA)
  ln_mlp_fused_kernel<<<NPTS / MT, 256, LDS_BYTES, stream>>>(
      x, acc, ln_g, ln_b, w1, b1, w2, b2, gamma, out);
}